// yoloLoss_79259326480918
// MI455X (gfx1250) — compile-verified
//
#include <hip/hip_runtime.h>
#include <cstdint>

// ---------------- problem constants ----------------
#define GRID_N   14
#define BOXN     2
#define LABELN   20
#define ALLBOX   10            // 5*BOXN
#define C_CH     30            // ALLBOX + LABELN
#define B_BATCH  4096
#define N_CELLS  (B_BATCH * GRID_N * GRID_N)   // 802816
#define CELLS_PER_BLOCK 256
#define THREADS  256
#define TILE_FLOATS (CELLS_PER_BLOCK * C_CH)   // 7680 floats per tensor tile
#define TILE_VEC4   (TILE_FLOATS / 4)          // 1920 float4 per tensor tile
#define NBLOCKS  (N_CELLS / CELLS_PER_BLOCK)   // 3136
#define CELL_SZ  (1.0f / 14.0f)

typedef __attribute__((ext_vector_type(2))) float v2f;
typedef __attribute__((ext_vector_type(8))) float v8f;

// ---------------- CDNA5 feature detection (compile-safe guards) -------------
#if defined(__gfx1250__) && __has_builtin(__builtin_amdgcn_global_load_async_to_lds_b128)
#define HAS_ASYNC_LDS 1
#else
#define HAS_ASYNC_LDS 0
#endif

#if defined(__gfx1250__) && __has_builtin(__builtin_amdgcn_wmma_f32_16x16x4_f32)
#define HAS_WMMA_F32X4 1
#else
#define HAS_WMMA_F32X4 0
#endif

// async DMA global -> LDS (16B per lane, coalesced). The builtin takes typed
// vector pointers: (addrspace(1) int4-vec*, addrspace(3) int4-vec*, imm, imm).
// Falls back to a plain vector copy if the builtin is unavailable.
#if HAS_ASYNC_LDS
typedef int v4i __attribute__((vector_size(4 * sizeof(int))));
#endif

__device__ __forceinline__ void copy16_g2lds(const float4* __restrict__ gsrc,
                                             float4* __restrict__ ldst) {
#if HAS_ASYNC_LDS
  __builtin_amdgcn_global_load_async_to_lds_b128(
      (__attribute__((address_space(1))) v4i*)(uintptr_t)gsrc,
      (__attribute__((address_space(3))) v4i*)(uintptr_t)ldst,
      0, 0);
#else
  *ldst = *gsrc;
#endif
}

__device__ __forceinline__ void wait_async_done() {
#if HAS_ASYNC_LDS
#if __has_builtin(__builtin_amdgcn_s_wait_asynccnt)
  __builtin_amdgcn_s_wait_asynccnt(0);
#else
  asm volatile("s_wait_asynccnt 0" ::: "memory");
#endif
#endif
}

// Wave32 sum-reduction. Primary path: one V_WMMA_F32_16X16X4_F32 with B=ones:
//   A[m][0]=v(lane m), A[m][2]=v(lane m+16)  ->  D[m][n] = v_m + v_{m+16}
// Each lane then owns 8 D rows; local add + one shfl_xor(16) yields the full
// 32-lane sum in every lane. Requires EXEC all-ones (call convergently).
__device__ __forceinline__ float wave_reduce_sum(float v) {
#if HAS_WMMA_F32X4
  v2f a; a[0] = v;    a[1] = 0.0f;
  v2f b; b[0] = 1.0f; b[1] = 1.0f;
  v8f c = {0.f, 0.f, 0.f, 0.f, 0.f, 0.f, 0.f, 0.f};
  v8f d = __builtin_amdgcn_wmma_f32_16x16x4_f32(
      /*neg_a=*/false, a, /*neg_b=*/false, b,
      /*c_mod=*/(short)0, c, /*reuse_a=*/false, /*reuse_b=*/false);
  float s = ((d[0] + d[1]) + (d[2] + d[3])) + ((d[4] + d[5]) + (d[6] + d[7]));
  s += __shfl_xor(s, 16, 32);
  return s;
#else
#pragma unroll
  for (int off = 16; off > 0; off >>= 1) v += __shfl_xor(v, off, 32);
  return v;
#endif
}

// -------------------- kernel 1: per-block partial loss ----------------------
__global__ __launch_bounds__(THREADS)
void yolo_loss_partial(const float* __restrict__ preds,
                       const float* __restrict__ truths,
                       float* __restrict__ partial) {
  __shared__ float lds_p[TILE_FLOATS];
  __shared__ float lds_t[TILE_FLOATS];
  __shared__ float red[THREADS / 32];

  const int tid  = threadIdx.x;
  const size_t base = (size_t)blockIdx.x * TILE_FLOATS;
  const float4* gp = (const float4*)(preds  + base);
  const float4* gt = (const float4*)(truths + base);
  float4* lp4 = (float4*)lds_p;
  float4* lt4 = (float4*)lds_t;

  // Stage 256 cells (7680 floats per tensor) via coalesced 16B async loads.
#pragma unroll
  for (int i = 0; i < 8; ++i) {
    const int idx = i * THREADS + tid;
    if (idx < TILE_VEC4) {
      copy16_g2lds(gp + idx, lp4 + idx);
      copy16_g2lds(gt + idx, lt4 + idx);
    }
  }
  wait_async_done();
  __syncthreads();

  // Each thread owns one cell; 30-dword stride is bank-conflict-free in wave32.
  float p[C_CH], t[C_CH];
  {
    const float2* sp = (const float2*)(lds_p + tid * C_CH);
    const float2* st = (const float2*)(lds_t + tid * C_CH);
#pragma unroll
    for (int i = 0; i < C_CH / 2; ++i) {
      float2 vp = sp[i]; p[2 * i] = vp.x; p[2 * i + 1] = vp.y;
      float2 vt = st[i]; t[2 * i] = vt.x; t[2 * i + 1] = vt.y;
    }
  }

  // ---- IoU of both predicted boxes vs the truth box ----
  const float tc0 = CELL_SZ * t[0], tc1 = CELL_SZ * t[1];
  const float th0 = 0.5f * t[2],    th1 = 0.5f * t[3];
  const float tl0 = tc0 - th0, tl1 = tc1 - th1;
  const float tr0 = tc0 + th0, tr1 = tc1 + th1;
  const float ta  = (tr0 - tl0) * (tr1 - tl1);

  float iou[BOXN], conf[BOXN];
#pragma unroll
  for (int b = 0; b < BOXN; ++b) {
    const int o = 5 * b;
    const float pc0 = CELL_SZ * p[o + 0], pc1 = CELL_SZ * p[o + 1];
    const float ph0 = 0.5f * p[o + 2],    ph1 = 0.5f * p[o + 3];
    const float pl0 = pc0 - ph0, pl1 = pc1 - ph1;
    const float pr0 = pc0 + ph0, pr1 = pc1 + ph1;
    const float lt0 = fmaxf(pl0, tl0), lt1 = fmaxf(pl1, tl1);
    const float rb0 = fminf(pr0, tr0), rb1 = fminf(pr1, tr1);
    const float w = fmaxf(rb0 - lt0, 0.0f), h = fmaxf(rb1 - lt1, 0.0f);
    const float inter = w * h;
    const float pa = (pr0 - pl0) * (pr1 - pl1);
    iou[b]  = inter / (pa + ta - inter);
    conf[b] = p[o + 4];
  }

  // ---- responsible-box selection (matches jnp argmax tie rules, BOXN=2) ----
  const float max_iou = fmaxf(iou[0], iou[1]);
  int resp;
  if (iou[0] == iou[1]) resp = (conf[0] >= conf[1]) ? 0 : 1;  // tie -> argmax conf
  else                  resp = (iou[0] >  iou[1]) ? 0 : 1;    // argmax iou
  const int ro = 5 * resp;

  // ---- per-cell loss ----
  float cell_loss;
  if (t[4] > 0.0f) {  // object present
    const float d0 = p[ro + 0] - t[0];
    const float d1 = p[ro + 1] - t[1];
    const float center = d0 * d0 + d1 * d1;
    const float s2 = sqrtf(p[ro + 2]) - sqrtf(t[2]);
    const float s3 = sqrtf(p[ro + 3]) - sqrtf(t[3]);
    const float size = s2 * s2 + s3 * s3;
    const float dc = p[ro + 4] - max_iou;
    const float conf_resp = dc * dc;
    const float cnr = conf[1 - resp];
    const float conf_noresp = cnr * cnr;
    float label = 0.0f;
#pragma unroll
    for (int k = ALLBOX; k < C_CH; ++k) {
      const float dl = p[k] - t[k];
      label += dl * dl;
    }
    cell_loss = 5.0f * (center + size) + conf_resp + 0.5f * conf_noresp + label;
  } else {            // no object: 0.5 * sum(conf^2) over both boxes
    cell_loss = 0.5f * (conf[0] * conf[0] + conf[1] * conf[1]);
  }

  // ---- block reduction: WMMA wave-sum, then cross-wave via LDS ----
  const float wsum = wave_reduce_sum(cell_loss);
  const int wid = tid >> 5, lane = tid & 31;
  if (lane == 0) red[wid] = wsum;
  __syncthreads();
  if (tid < 32) {
    const float v = (lane < THREADS / 32) ? red[lane] : 0.0f;
    const float bsum = wave_reduce_sum(v);
    if (lane == 0) partial[blockIdx.x] = bsum;
  }
}

// -------------------- kernel 2: finalize (single block) ---------------------
__global__ __launch_bounds__(THREADS)
void yolo_loss_finalize(const float* __restrict__ partial, int n,
                        float* __restrict__ out) {
  __shared__ float red[THREADS / 32];
  const int tid = threadIdx.x;
  float acc = 0.0f;
  for (int i = tid; i < n; i += THREADS) acc += partial[i];
  const float wsum = wave_reduce_sum(acc);
  const int wid = tid >> 5, lane = tid & 31;
  if (lane == 0) red[wid] = wsum;
  __syncthreads();
  if (tid < 32) {
    const float v = (lane < THREADS / 32) ? red[lane] : 0.0f;
    const float bsum = wave_reduce_sum(v);
    if (lane == 0) out[0] = bsum * (1.0f / (float)B_BATCH);
  }
}

// ------------------------------- launcher -----------------------------------
extern "C" void kernel_launch(void* const* d_in, const int* in_sizes, int n_in,
                              void* d_out, int out_size, void* d_ws, size_t ws_size,
                              hipStream_t stream) {
  const float* preds  = (const float*)d_in[0];
  const float* truths = (const float*)d_in[1];
  float* out     = (float*)d_out;
  float* partial = (float*)d_ws;   // NBLOCKS floats (12.5 KB) of scratch

  yolo_loss_partial<<<NBLOCKS, THREADS, 0, stream>>>(preds, truths, partial);
  yolo_loss_finalize<<<1, THREADS, 0, stream>>>(partial, NBLOCKS, out);
}